// SpatialCorrelationSampler_44255343018948
// MI455X (gfx1250) — compile-verified
//
#include <hip/hip_runtime.h>
#include <stdint.h>

// ---------------------------------------------------------------------------
// SpatialCorrelationSampler (kernel=1, patch=2, stride=2) on MI455X / gfx1250.
// Bandwidth-bound streaming reduction: 268 MB in, 0.5 MB out, 67 MFLOP
// => ~11.5 us roofline at 23.3 TB/s; WMMA cannot help at 0.25 FLOP/byte fp32.
// Strategy: gfx1250 async global->LDS (ASYNCcnt) double-buffered streaming,
// 16 outstanding b128 DMAs per wave for latency coverage, fully-coalesced
// 512 B row chunks, fp32 VALU FMA, deterministic LDS tree reduce.
// ---------------------------------------------------------------------------

#if __has_builtin(__builtin_amdgcn_global_load_async_to_lds_b128)
#define USE_ASYNC_LDS 1
#else
#define USE_ASYNC_LDS 0
#endif

// Builtin signature (clang gfx1250): (v4i addrspace(1)*, v4i addrspace(3)*,
// imm int offset, imm int cpol)  -- "vV4i*1V4i*3IiIi"
typedef int v4i_t __attribute__((vector_size(16)));
typedef __attribute__((address_space(1))) v4i_t* gv4i_p;  // global
typedef __attribute__((address_space(3))) v4i_t* lv4i_p;  // LDS

namespace {
constexpr int kB = 8, kC = 256, kH = 128, kW = 128, kOH = 64, kOW = 64;
constexpr int kChPerStage = 16;                   // channels staged per step
constexpr int kStages = kC / kChPerStage;         // 16
constexpr int kLoadsPerThread = kChPerStage * 4 * (kW / 4) / 256;  // 8
constexpr int kBufFloats = kChPerStage * 4 * kW;  // 16 ch * 4 rows * 128 = 8192 (32 KB)
}  // namespace

template <int N>
__device__ __forceinline__ void wait_async() {
#if USE_ASYNC_LDS
#if __has_builtin(__builtin_amdgcn_s_wait_asynccnt)
  __builtin_amdgcn_s_wait_asynccnt(N);
#else
  asm volatile("s_wait_asynccnt %0" ::"i"(N) : "memory");
#endif
#endif
}

// Stage 16 channels (rows 2*oy, 2*oy+1 of both inputs) into LDS.
// 2048 16-byte chunks per stage / 256 threads = 8 async b128 loads per thread.
// Within a wave, 32 lanes cover one contiguous 512 B row -> fully coalesced.
__device__ __forceinline__ void issue_stage(const float* __restrict__ A,
                                            const float* __restrict__ B2,
                                            float* lds, size_t abase, int s,
                                            int buf, int t) {
#pragma unroll
  for (int i = 0; i < kLoadsPerThread; ++i) {
    const int id = t + 256 * i;   // 0..2047
    const int ch = id >> 7;       // 0..15  channel within stage
    const int rem = id & 127;
    const int rowsel = rem >> 5;  // 0: A row0, 1: A row1, 2: B row0, 3: B row1
    const int col = rem & 31;     // 16-byte chunk within a 512 B row
    const int channel = s * kChPerStage + ch;
    const float* src = (rowsel < 2) ? A : B2;
    const size_t g = abase + (size_t)channel * (kH * kW) +
                     (size_t)(rowsel & 1) * kW + (size_t)col * 4;
    const int l = buf * kBufFloats + ch * 512 + rowsel * 128 + col * 4;
#if USE_ASYNC_LDS
    float* gs = const_cast<float*>(src + g);
    __builtin_amdgcn_global_load_async_to_lds_b128((gv4i_p)gs,
                                                   (lv4i_p)(lds + l), 0, 0);
#else
    const float4 v = *(const float4*)(src + g);
    *(float4*)(lds + l) = v;
#endif
  }
}

__global__ __launch_bounds__(256) void corr_patch_kernel(
    const float* __restrict__ A, const float* __restrict__ B2,
    float* __restrict__ out) {
  __shared__ float lds[2 * kBufFloats];  // 64 KB double buffer

  const int wg = blockIdx.x;  // 0..511 = b*64 + oy
  const int b = wg >> 6;
  const int oy = wg & 63;
  const int t = (int)threadIdx.x;  // 0..255
  const int ox = t & 63;           // output column
  const int grp = t >> 6;          // 0..3 channel group (4 ch/stage each)

  // Zero-fill the three patch slots the reference never writes:
  // out[b, ph, pw, oy, :] for (ph,pw) in {(0,1),(1,0),(1,1)} -> grp = 1,2,3.
  if (grp > 0)
    out[(size_t)b * 16384 + (size_t)grp * 4096 + (size_t)oy * 64 + ox] = 0.0f;

  // Base (in floats) of element [b, c=0, 2*oy, 0].
  const size_t abase = (size_t)b * kC * kH * kW + (size_t)(2 * oy) * kW;

  float acc = 0.0f;

  issue_stage(A, B2, lds, abase, /*s=*/0, /*buf=*/0, t);
  for (int s = 0; s < kStages; ++s) {
    const int buf = s & 1;
    if (s + 1 < kStages) {
      issue_stage(A, B2, lds, abase, s + 1, buf ^ 1, t);
      wait_async<kLoadsPerThread>();  // in-order completion: stage s resident
    } else {
      wait_async<0>();
    }
    __syncthreads();  // all waves' DMA for stage s landed in LDS

    // Each thread: 4 channels of this stage, one output column (2x2 block).
#pragma unroll
    for (int j = 0; j < 4; ++j) {
      const float* base = lds + buf * kBufFloats + (grp * 4 + j) * 512 + 2 * ox;
      const float2 a0 = *(const float2*)(base);        // A row 2*oy
      const float2 a1 = *(const float2*)(base + 128);  // A row 2*oy+1
      const float2 b0 = *(const float2*)(base + 256);  // B row 2*oy
      const float2 b1 = *(const float2*)(base + 384);  // B row 2*oy+1
      acc += a0.x * b0.x + a0.y * b0.y + a1.x * b1.x + a1.y * b1.y;
    }
    __syncthreads();  // buffer may be DMA-overwritten next iteration
  }

  // Deterministic fixed-order reduction of the 4 channel-group partials.
  lds[t] = acc;  // slot = grp*64 + ox == t
  __syncthreads();
  if (t < 64) {
    const float r = lds[t] + lds[t + 64] + lds[t + 128] + lds[t + 192];
    out[(size_t)b * 16384 + (size_t)oy * 64 + t] = r;  // patch slot (0,0)
  }
}

extern "C" void kernel_launch(void* const* d_in, const int* in_sizes, int n_in,
                              void* d_out, int out_size, void* d_ws,
                              size_t ws_size, hipStream_t stream) {
  (void)in_sizes;
  (void)n_in;
  (void)d_ws;
  (void)ws_size;
  (void)out_size;
  const float* A = (const float*)d_in[0];
  const float* B2 = (const float*)d_in[1];
  float* out = (float*)d_out;
  corr_patch_kernel<<<dim3(kB * kOH), dim3(256), 0, stream>>>(A, B2, out);
}